// SparseAttentionLayer_3788161155598
// MI455X (gfx1250) — compile-verified
//
#include <hip/hip_runtime.h>
#include <hip/hip_bf16.h>

#define DIM 128

typedef __attribute__((ext_vector_type(2))) float v2f;
typedef __attribute__((ext_vector_type(8))) float v8f;

// ---- monotonic uint encoding of float for atomicMax-based segment max ----
__device__ __forceinline__ unsigned f2key(float x) {
    unsigned b = __float_as_uint(x);
    return (b & 0x80000000u) ? ~b : (b | 0x80000000u);
}
__device__ __forceinline__ float key2f(unsigned k) {
    unsigned b = (k & 0x80000000u) ? (k ^ 0x80000000u) : ~k;
    return __uint_as_float(b);
}

// ---- init: zero output, segment sums, segment max keys ----
__global__ void init_kernel(float* __restrict__ out, float* __restrict__ seg_sum,
                            unsigned* __restrict__ seg_key, long long n_out, int n_ent) {
    long long i = (long long)blockIdx.x * blockDim.x + threadIdx.x;
    long long stride = (long long)gridDim.x * blockDim.x;
    for (; i < n_out; i += stride) {
        out[i] = 0.0f;
        if (i < n_ent) { seg_sum[i] = 0.0f; seg_key[i] = 0u; }
    }
}

// ---- rel_score = leaky_relu(relations @ w + b) via V_WMMA_F32_16X16X4_F32 ----
// One wave32 per 16-row tile. A = 16x4 chunk of relations, B[k][n] = w[k]
// broadcast across N, so D[m][n] = sum_k rel[m,k]*w[k] for every n.
__global__ void rel_score_kernel(const float* __restrict__ rel,
                                 const float* __restrict__ w,
                                 const float* __restrict__ bias,
                                 float* __restrict__ rel_score, int n_rel) {
    const int lane  = threadIdx.x;          // blockDim.x == 32
    const int m     = lane & 15;            // row within tile
    const int khalf = lane >> 4;            // 0 -> K{0,1}, 1 -> K{2,3}
    const int tile  = blockIdx.x;
    int row = tile * 16 + m;
    int rowc = row < n_rel ? row : (n_rel - 1);   // clamp: keep EXEC all-ones
    const float* arow = rel + (long long)rowc * DIM;

    v8f c = {};
    for (int kb = 0; kb < DIM; kb += 4) {
        int k0 = kb + 2 * khalf;
        v2f a, b;
        a.x = arow[k0];     a.y = arow[k0 + 1];
        b.x = w[k0];        b.y = w[k0 + 1];
        c = __builtin_amdgcn_wmma_f32_16x16x4_f32(
                false, a, false, b, (short)0, c, false, false);
    }
    // D layout: lane 0 -> (M=0..7, N=0) in c[0..7]; lane 16 -> (M=8..15, N=0)
    float bv = bias[0];
    if (lane == 0 || lane == 16) {
        int mbase = tile * 16 + (lane == 16 ? 8 : 0);
        #pragma unroll
        for (int r = 0; r < 8; ++r) {
            int orow = mbase + r;
            if (orow < n_rel) {
                float s = c[r] + bv;
                rel_score[orow] = s > 0.0f ? s : 0.01f * s;   // leaky_relu
            }
        }
    }
}

// ---- pass 1: segment max over edges (atomicMax on monotone key) ----
__global__ void seg_max_kernel(const float* __restrict__ rel_score,
                               const int* __restrict__ h_idx,
                               const int* __restrict__ r_idx,
                               unsigned* __restrict__ seg_key, int n_edge) {
    int i = blockIdx.x * blockDim.x + threadIdx.x;
    int stride = gridDim.x * blockDim.x;
    for (; i < n_edge; i += stride) {
        float wv = rel_score[r_idx[i]];
        atomicMax(&seg_key[h_idx[i]], f2key(wv));
    }
}

// ---- pass 2: segment sum of exp(w - max) ----
__global__ void seg_sum_kernel(const float* __restrict__ rel_score,
                               const int* __restrict__ h_idx,
                               const int* __restrict__ r_idx,
                               const unsigned* __restrict__ seg_key,
                               float* __restrict__ seg_sum, int n_edge) {
    int i = blockIdx.x * blockDim.x + threadIdx.x;
    int stride = gridDim.x * blockDim.x;
    for (; i < n_edge; i += stride) {
        int h = h_idx[i];
        float wv = rel_score[r_idx[i]];
        float mv = key2f(seg_key[h]);
        atomicAdd(&seg_sum[h], expf(wv - mv));
    }
}

// ---- pass 3: wave32 per edge: out[h] += coef * entities[t] ----
// Each lane owns a 16B slice of the 512B row: global_load_b128 gather +
// 4x global_atomic_add_f32, with a global_prefetch of the next edge's row.
__global__ void scatter_kernel(const float* __restrict__ ent,
                               const float* __restrict__ rel_score,
                               const unsigned* __restrict__ seg_key,
                               const float* __restrict__ seg_sum,
                               const int* __restrict__ h_idx,
                               const int* __restrict__ r_idx,
                               const int* __restrict__ t_idx,
                               float* __restrict__ out, int n_edge) {
    const int lane = threadIdx.x & 31;
    const int wid  = blockIdx.x * (blockDim.x >> 5) + (threadIdx.x >> 5);
    const int nw   = gridDim.x * (blockDim.x >> 5);
    for (int e = wid; e < n_edge; e += nw) {
        int h = h_idx[e], r = r_idx[e], t = t_idx[e];
        int en = e + nw;
        if (en < n_edge) {   // prefetch next iteration's entity row into L0/L2
            int tn = t_idx[en];
            __builtin_prefetch(&ent[(long long)tn * DIM + lane * 4], 0, 0);
        }
        float wv   = rel_score[r];
        float mv   = key2f(seg_key[h]);
        float coef = expf(wv - mv) / seg_sum[h];
        const float4 v = *(const float4*)(ent + (long long)t * DIM + lane * 4);
        float* o = out + (long long)h * DIM + lane * 4;
        atomicAdd(o + 0, coef * v.x);
        atomicAdd(o + 1, coef * v.y);
        atomicAdd(o + 2, coef * v.z);
        atomicAdd(o + 3, coef * v.w);
    }
}

// ---- final relu ----
__global__ void relu_kernel(float* __restrict__ out, long long n) {
    long long i = (long long)blockIdx.x * blockDim.x + threadIdx.x;
    long long stride = (long long)gridDim.x * blockDim.x;
    for (; i < n; i += stride) {
        float v = out[i];
        out[i] = v > 0.0f ? v : 0.0f;
    }
}

extern "C" void kernel_launch(void* const* d_in, const int* in_sizes, int n_in,
                              void* d_out, int out_size, void* d_ws, size_t ws_size,
                              hipStream_t stream) {
    const float* entities  = (const float*)d_in[0];
    const float* relations = (const float*)d_in[1];
    const float* w_trans   = (const float*)d_in[2];
    const float* b_trans   = (const float*)d_in[3];
    const int*   h_idx     = (const int*)d_in[4];
    const int*   r_idx     = (const int*)d_in[5];
    const int*   t_idx     = (const int*)d_in[6];
    float* out = (float*)d_out;

    const int n_ent  = in_sizes[0] / DIM;
    const int n_rel  = in_sizes[1] / DIM;
    const int n_edge = in_sizes[4];
    const long long n_out = (long long)n_ent * DIM;

    // workspace layout: rel_score[512] | seg_key[n_ent] | seg_sum[n_ent]
    float*    rel_score = (float*)d_ws;
    unsigned* seg_key   = (unsigned*)((char*)d_ws + 512 * sizeof(float));
    float*    seg_sum   = (float*)((char*)d_ws + 512 * sizeof(float)
                                   + (size_t)n_ent * sizeof(unsigned));

    // 1) zero out/sums/keys
    init_kernel<<<2048, 256, 0, stream>>>(out, seg_sum, seg_key, n_out, n_ent);

    // 2) rel_score via WMMA (wave32 per 16-row tile)
    int n_tiles = (n_rel + 15) / 16;
    rel_score_kernel<<<n_tiles, 32, 0, stream>>>(relations, w_trans, b_trans,
                                                 rel_score, n_rel);

    // 3) segment max
    seg_max_kernel<<<(n_edge + 255) / 256, 256, 0, stream>>>(
        rel_score, h_idx, r_idx, seg_key, n_edge);

    // 4) segment sum of exp
    seg_sum_kernel<<<(n_edge + 255) / 256, 256, 0, stream>>>(
        rel_score, h_idx, r_idx, seg_key, seg_sum, n_edge);

    // 5) scatter: wave per edge
    scatter_kernel<<<6144, 256, 0, stream>>>(
        entities, rel_score, seg_key, seg_sum, h_idx, r_idx, t_idx, out, n_edge);

    // 6) relu
    relu_kernel<<<2048, 256, 0, stream>>>(out, n_out);
}